// Encoding_61933428412497
// MI455X (gfx1250) — compile-verified
//
#include <hip/hip_runtime.h>
#include <stdint.h>

typedef __attribute__((ext_vector_type(2))) float v2f;
typedef __attribute__((ext_vector_type(8))) float v8f;

#define N_ 32
#define D_ 128
#define L_ 3136
#define K_ 64

// CDNA5 async global->LDS DMA (ASYNCcnt-tracked). %0 = per-lane LDS byte
// address (VGPR), %1 = per-lane global address (VGPR pair), no SADDR.
__device__ __forceinline__ void async_ld_b64(uint32_t lds_addr, const float* gptr) {
    asm volatile("global_load_async_to_lds_b64 %0, %1, off"
                 :: "v"(lds_addr), "v"(gptr) : "memory");
}
__device__ __forceinline__ void async_ld_b128(uint32_t lds_addr, const float* gptr) {
    asm volatile("global_load_async_to_lds_b128 %0, %1, off"
                 :: "v"(lds_addr), "v"(gptr) : "memory");
}

__global__ __launch_bounds__(256)
void enc_dist_kernel(const float* __restrict__ x,
                     const float* __restrict__ w,
                     const float* __restrict__ scale,
                     float* __restrict__ out) {
    // Weight: paired-K layout, element (k,row) at sW[(k>>1)*128 + row*2 + (k&1)]
    // (conflict-free A fetch; global pairs map to contiguous 8B -> async b64).
    __shared__ float sW[64 * 128];   // 32 KB
    // X tile: plain [d][col] layout (global rows map to contiguous LDS -> async b128).
    __shared__ float sX[128 * 64];   // 32 KB
    __shared__ float sCsq[64];       // ||w_k||^2
    __shared__ float sXsq[64];       // ||x_l||^2

    const int tid  = threadIdx.x;
    const int n    = blockIdx.y;
    const int col0 = blockIdx.x * 64;

    const float* xb = x + (size_t)n * D_ * L_ + col0;
    const uint32_t sWb = (uint32_t)(uintptr_t)(&sW[0]);
    const uint32_t sXb = (uint32_t)(uintptr_t)(&sX[0]);

    // ---- Async DMA staging: weight 64x128 -> paired layout (4096 b64 xfers) ----
#pragma unroll
    for (int j = 0; j < 16; ++j) {
        int i   = tid + j * 256;            // 0..4095
        int row = i >> 6;
        int p   = i & 63;                   // lanes -> consecutive p: coalesced global
        async_ld_b64(sWb + (uint32_t)(p * 128 + row * 2) * 4u,
                     w + row * 128 + 2 * p);
    }
    // ---- Async DMA staging: x tile 128x64 -> plain layout (2048 b128 xfers) ----
#pragma unroll
    for (int j = 0; j < 8; ++j) {
        int i  = tid + j * 256;             // 0..2047
        int d  = i >> 4;
        int c4 = (i & 15) * 4;              // lanes -> consecutive 16B: coalesced
        async_ld_b128(sXb + (uint32_t)(d * 64 + c4) * 4u,
                      xb + (size_t)d * L_ + c4);
    }
    asm volatile("s_wait_asynccnt 0x0" ::: "memory");
    __syncthreads();

    // ---- Per-row / per-column squared norms (once per block) ----
    if (tid < 64) {
        int row2 = tid * 2;
        float s = 0.f;
        for (int p = 0; p < 64; ++p) {
            float e0 = sW[p * 128 + row2];
            float e1 = sW[p * 128 + row2 + 1];
            s += e0 * e0 + e1 * e1;
        }
        sCsq[tid] = s;
    } else if (tid < 128) {
        int col = tid - 64;
        float s = 0.f;
        for (int d = 0; d < 128; ++d) {
            float e = sX[d * 64 + col];
            s += e * e;
        }
        sXsq[col] = s;
    }
    __syncthreads();

    // ---- WMMA main loop: wave w -> column tile (w&3), row tiles (w>>2), (w>>2)+2 ----
    const int wave = tid >> 5;                  // wave32
    const int lane = tid & 31;
    const int lm   = lane & 15;
    const int h    = lane >> 4;                 // lane-half: K+0/K+2 (A,B), M+8 (C/D)
    const int c    = wave & 3;
    const int r0   = wave >> 2;                 // 0..1
    const int r1   = r0 + 2;                    // 2..3

    v8f acc0 = {};
    v8f acc1 = {};
#pragma unroll 8
    for (int t = 0; t < 32; ++t) {              // k0 = 4*t, full D=128 from LDS
        int p = 2 * t + h;                      // weight k-pair index: d = 4t+2h, +1
        v2f a0 = *(const v2f*)(&sW[p * 128 + (r0 * 16 + lm) * 2]);
        v2f a1 = *(const v2f*)(&sW[p * 128 + (r1 * 16 + lm) * 2]);
        int bbase = (4 * t + 2 * h) * 64 + c * 16 + lm;
        v2f b;
        b.x = sX[bbase];                        // d = 4t+2h   (VGPR0: K0/K2)
        b.y = sX[bbase + 64];                   // d = 4t+2h+1 (VGPR1: K1/K3)
        acc0 = __builtin_amdgcn_wmma_f32_16x16x4_f32(false, a0, false, b,
                                                     (short)0, acc0, false, false);
        acc1 = __builtin_amdgcn_wmma_f32_16x16x4_f32(false, a1, false, b,
                                                     (short)0, acc1, false, false);
    }

    // ---- Fused epilogue: (||x||^2 + ||c||^2 - 2*xc) / scale^2 ----
    float sv  = scale[0];
    float inv = 1.0f / (sv * sv);
    int   l   = col0 + c * 16 + lm;             // output column (N = lm, both halves)
    float xs  = sXsq[c * 16 + lm];
    size_t outn = (size_t)n * K_ * L_;
#pragma unroll
    for (int j = 0; j < 8; ++j) {
        int M  = j + h * 8;                     // C/D: VGPR j -> rows j / j+8
        int k0 = r0 * 16 + M;
        int k1 = r1 * 16 + M;
        out[outn + (size_t)k0 * L_ + l] = (xs + sCsq[k0] - 2.0f * acc0[j]) * inv;
        out[outn + (size_t)k1 * L_ + l] = (xs + sCsq[k1] - 2.0f * acc1[j]) * inv;
    }
}

extern "C" void kernel_launch(void* const* d_in, const int* in_sizes, int n_in,
                              void* d_out, int out_size, void* d_ws, size_t ws_size,
                              hipStream_t stream) {
    const float* x     = (const float*)d_in[0];
    const float* wgt   = (const float*)d_in[1];
    const float* scale = (const float*)d_in[2];
    float* out = (float*)d_out;
    dim3 grid(L_ / 64, N_);   // 49 x 32 blocks
    enc_dist_kernel<<<grid, 256, 0, stream>>>(x, wgt, scale, out);
}